// SCGRN_46110768890277
// MI455X (gfx1250) — compile-verified
//
#include <hip/hip_runtime.h>
#include <math.h>

#define Nn 1024
#define Mm 4
#define Ff 64
#define Ll 4
#define Kk 32

typedef float v2f __attribute__((ext_vector_type(2)));
typedef float v8f __attribute__((ext_vector_type(8)));

__device__ __forceinline__ float lrelu(float x) { return x > 0.0f ? x : 0.01f * x; }

__device__ __forceinline__ float dot64(const float* __restrict__ a, const float* __restrict__ b) {
    float s = 0.0f;
#pragma unroll
    for (int t = 0; t < 64; ++t) s = fmaf(a[t], b[t], s);
    return s;
}

// ---------------- Kernel 1: enc, E2 = enc@Wq2^T, C = E2@mk^T, MVP = mv@Wp_w^T ----------------
__global__ void scgrn_k1_small(const float* __restrict__ Wq, const float* __restrict__ mk,
                               const float* __restrict__ mv, const float* __restrict__ Wp_w,
                               float* __restrict__ Cws, float* __restrict__ MVP) {
    __shared__ float enc[Mm * Ff];
    __shared__ float E2[Mm * Ff];
    const int tid = threadIdx.x;
    {
        const int m = tid >> 6, f = tid & 63;
        const int t = f >> 1;
        // div[t] = exp(-(2t)*ln(10000)/64)
        float dv = __expf((float)(2 * t) * (-9.210340371976184f / 64.0f));
        float v = (float)m * dv;
        enc[tid] = (f & 1) ? __cosf(v) : __sinf(v);
    }
    __syncthreads();
    {
        const int m = tid >> 6, f = tid & 63;
        E2[tid] = dot64(enc + m * 64, Wq + f * 128 + 64);
    }
    __syncthreads();
    if (tid < 128) {
        const int m = tid >> 5, k = tid & 31;
        Cws[m * Kk + k] = dot64(E2 + m * 64, mk + k * 64);
    } else {
        const int idx = tid - 128;
        const int k = idx >> 2, l = idx & 3;
        MVP[k * Ll + l] = dot64(mv + k * 64, Wp_w + l * 64);
    }
}

// ---------------- Kernel 2: per-node precomputes ----------------
// nf_sum, q1 = h_i@Wq1^T, s2 = nf_sum@Wq2^T  (kept in LDS)
// outputs: A=q1@mk^T, B=s2@mk^T, h_all[l]=h_i@Wgat[l]^T, f1, f2
__global__ void scgrn_k2_node(const float* __restrict__ nf, const float* __restrict__ Wq,
                              const float* __restrict__ mk, const float* __restrict__ Wgat,
                              const float* __restrict__ agat,
                              float* __restrict__ Aws, float* __restrict__ Bws,
                              float* __restrict__ hall, float* __restrict__ f1,
                              float* __restrict__ f2) {
    __shared__ float hi[64], nfs[64], q1s[64], s2s[64], hl[256];
    const int i = blockIdx.x, tid = threadIdx.x;
    if (tid < 64) {
        const float* base = nf + (size_t)i * (Mm * Ff) + tid;
        float h0 = base[0];
        hi[tid] = h0;
        nfs[tid] = h0 + base[64] + base[128] + base[192];
    }
    __syncthreads();
    {
        const int l = tid >> 6, d = tid & 63;
        float v = dot64(hi, Wgat + (size_t)(l * 64 + d) * 64);
        hl[tid] = v;
        hall[((size_t)l * Nn + i) * 64 + d] = v;
    }
    if (tid < 64) {
        q1s[tid] = dot64(hi, Wq + tid * 128);
    } else if (tid < 128) {
        const int f = tid - 64;
        s2s[f] = dot64(nfs, Wq + f * 128 + 64);
    }
    __syncthreads();
    if (tid < 32) {
        Aws[i * Kk + tid] = dot64(q1s, mk + tid * 64);
    } else if (tid < 64) {
        const int k = tid - 32;
        Bws[i * Kk + k] = dot64(s2s, mk + k * 64);
    } else if (tid < 68) {
        const int l = tid - 64;
        f1[l * Nn + i] = dot64(hl + l * 64, agat + l * 128);
    } else if (tid < 72) {
        const int l = tid - 68;
        f2[l * Nn + i] = dot64(hl + l * 64, agat + l * 128 + 64);
    }
}

// ---------------- Kernel 3: fused pair pass: mask, 32-way softmax, p ----------------
__global__ void scgrn_k3_pairs(const float* __restrict__ R, const float* __restrict__ Aws,
                               const float* __restrict__ Bws, const float* __restrict__ Cws,
                               const float* __restrict__ MVP, const float* __restrict__ Wp_b,
                               float* __restrict__ outp, unsigned char* __restrict__ maskb) {
    __shared__ float As[32], Cs[128], Ms[128], bs[4];
    const int i = blockIdx.y;
    const int tid = threadIdx.x;
    const int j = blockIdx.x * 256 + tid;
    if (tid < 32) As[tid] = Aws[i * 32 + tid];
    else if (tid < 160) Cs[tid - 32] = Cws[tid - 32];
    else if (tid < 288) Ms[tid - 160] = MVP[tid - 160];
    else if (tid < 292) bs[tid - 288] = Wp_b[tid - 288];
    __syncthreads();

    const float4 r = *(const float4*)(R + ((size_t)i * Nn + j) * 4);
    const bool mask = (r.x != 0.0f) | (r.y != 0.0f) | (r.z != 0.0f) | (r.w != 0.0f);

    float sc[32];
    float smax = -3.4e38f;
    const float* __restrict__ Bj = Bws + (size_t)j * 32;
#pragma unroll
    for (int k = 0; k < 32; ++k) {
        float s = As[k] + Bj[k];
        s = fmaf(r.x, Cs[k], s);
        s = fmaf(r.y, Cs[32 + k], s);
        s = fmaf(r.z, Cs[64 + k], s);
        s = fmaf(r.w, Cs[96 + k], s);
        s *= 0.125f;  // 1/sqrt(F)
        sc[k] = s;
        smax = fmaxf(smax, s);
    }
    float ssum = 0.0f;
#pragma unroll
    for (int k = 0; k < 32; ++k) { sc[k] = __expf(sc[k] - smax); ssum += sc[k]; }
    const float inv = 1.0f / ssum;

    float p0 = bs[0], p1 = bs[1], p2 = bs[2], p3 = bs[3];
#pragma unroll
    for (int k = 0; k < 32; ++k) {
        const float a = sc[k] * inv;
        p0 = fmaf(a, Ms[k * 4 + 0], p0);
        p1 = fmaf(a, Ms[k * 4 + 1], p1);
        p2 = fmaf(a, Ms[k * 4 + 2], p2);
        p3 = fmaf(a, Ms[k * 4 + 3], p3);
    }
    float4 po;
    if (mask) {
        const float pm = fmaxf(fmaxf(p0, p1), fmaxf(p2, p3));
        float e0 = __expf(p0 - pm), e1 = __expf(p1 - pm), e2 = __expf(p2 - pm), e3 = __expf(p3 - pm);
        const float pi = 1.0f / (e0 + e1 + e2 + e3);
        po.x = e0 * pi; po.y = e1 * pi; po.z = e2 * pi; po.w = e3 * pi;
    } else {
        po.x = 0.0f; po.y = 0.0f; po.z = 0.0f; po.w = 0.0f;
    }
    *(float4*)(outp + ((size_t)i * Nn + j) * 4) = po;
    maskb[(size_t)i * Nn + j] = mask ? 1 : 0;
}

// ---------------- Kernel 4: masked row reductions for GAT softmax ----------------
__global__ void scgrn_k4_rowred(const unsigned char* __restrict__ maskb,
                                const float* __restrict__ f1, const float* __restrict__ f2,
                                float* __restrict__ rowmax, float* __restrict__ rowinv) {
    __shared__ float sred[256];
    const int i = blockIdx.x, tid = threadIdx.x;
    const float NINF = -__builtin_inff();
    float f1l[4];
#pragma unroll
    for (int l = 0; l < 4; ++l) f1l[l] = f1[l * Nn + i];
    float vals[4][4];
    float tmax[4];
#pragma unroll
    for (int l = 0; l < 4; ++l) tmax[l] = NINF;
#pragma unroll
    for (int it = 0; it < 4; ++it) {
        const int j = it * 256 + tid;
        const unsigned char m = maskb[(size_t)i * Nn + j];
#pragma unroll
        for (int l = 0; l < 4; ++l) {
            float x = lrelu(f1l[l] + f2[l * Nn + j]);
            x = m ? x : NINF;
            vals[l][it] = x;
            tmax[l] = fmaxf(tmax[l], x);
        }
    }
    for (int l = 0; l < 4; ++l) {
        sred[tid] = tmax[l];
        __syncthreads();
        for (int s = 128; s > 0; s >>= 1) {
            if (tid < s) sred[tid] = fmaxf(sred[tid], sred[tid + s]);
            __syncthreads();
        }
        const float mx = sred[0];
        __syncthreads();
        float ts = 0.0f;
#pragma unroll
        for (int it = 0; it < 4; ++it) {
            const float v = vals[l][it];
            if (v > NINF) ts += __expf(v - mx);
        }
        sred[tid] = ts;
        __syncthreads();
        for (int s = 128; s > 0; s >>= 1) {
            if (tid < s) sred[tid] += sred[tid + s];
            __syncthreads();
        }
        const float tot = sred[0];
        __syncthreads();
        if (tid == 0) {
            rowmax[l * Nn + i] = mx;
            rowinv[l * Nn + i] = tot > 0.0f ? 1.0f / tot : 0.0f;
        }
    }
}

// ---------------- Kernel 5: v[i,d] = sum_{l,j} (p*alpha) * h_all via V_WMMA_F32_16X16X4_F32 ----------------
// One wave per 16x16 output tile. A = G[16i x 4j] fp32, B = h_all[4j x 16d] fp32, K=4 per WMMA.
__global__ void scgrn_k5_vgemm(const float* __restrict__ pbuf, const float* __restrict__ hall,
                               const float* __restrict__ f1, const float* __restrict__ f2,
                               const float* __restrict__ rowmax, const float* __restrict__ rowinv,
                               float* __restrict__ vout) {
    const int lane = threadIdx.x;
    const int lane16 = lane & 15;
    const int half = lane >> 4;
    const int iblk = (blockIdx.x >> 2) * 16;
    const int dblk = (blockIdx.x & 3) * 16;
    const int irow = iblk + lane16;  // A-matrix row M for this lane
    const int dcol = dblk + lane16;  // B-matrix col N for this lane
    v8f acc = {0.0f, 0.0f, 0.0f, 0.0f, 0.0f, 0.0f, 0.0f, 0.0f};
#pragma unroll 1
    for (int l = 0; l < 4; ++l) {
        const float fi = f1[l * Nn + irow];
        const float rmax = rowmax[l * Nn + irow];
        const float rinv = rowinv[l * Nn + irow];
        const float* __restrict__ f2l = f2 + l * Nn;
        const float* __restrict__ hl = hall + (size_t)l * Nn * 64;
        const float* __restrict__ prow = pbuf + (size_t)irow * Nn * 4 + l;
#pragma unroll 2
        for (int j0 = 0; j0 < Nn; j0 += 4) {
            const int ja = j0 + 2 * half;  // this lane holds K rows {ja, ja+1}
            const float x0 = lrelu(fi + f2l[ja]);
            const float x1 = lrelu(fi + f2l[ja + 1]);
            // alpha = exp(lrelu(f1+f2) - rowmax) / rowsum (0 if row fully masked);
            // p is already 0 on unmasked (i,j), so G = p * alpha needs no mask read.
            const float e0 = (rinv > 0.0f) ? __expf(x0 - rmax) * rinv : 0.0f;
            const float e1 = (rinv > 0.0f) ? __expf(x1 - rmax) * rinv : 0.0f;
            v2f a;
            a.x = prow[(size_t)ja * 4] * e0;
            a.y = prow[(size_t)(ja + 1) * 4] * e1;
            v2f b;
            b.x = hl[(size_t)ja * 64 + dcol];
            b.y = hl[(size_t)(ja + 1) * 64 + dcol];
            acc = __builtin_amdgcn_wmma_f32_16x16x4_f32(false, a, false, b, (short)0, acc,
                                                        false, false);
        }
    }
    // D layout: lanes 0-15 -> M=r, lanes 16-31 -> M=8+r ; N = lane16
#pragma unroll
    for (int r = 0; r < 8; ++r) {
        const int iout = iblk + half * 8 + r;
        vout[(size_t)iout * 64 + dcol] = lrelu(acc[r]);
    }
}

extern "C" void kernel_launch(void* const* d_in, const int* in_sizes, int n_in,
                              void* d_out, int out_size, void* d_ws, size_t ws_size,
                              hipStream_t stream) {
    const float* nf   = (const float*)d_in[0];  // (1024,4,64)
    const float* R    = (const float*)d_in[1];  // (1024,1024,4)
    const float* Wq   = (const float*)d_in[2];  // (64,128)
    const float* mk   = (const float*)d_in[3];  // (32,64)
    const float* mv   = (const float*)d_in[4];  // (32,64)
    const float* Wp_w = (const float*)d_in[5];  // (4,64)
    const float* Wp_b = (const float*)d_in[6];  // (4,)
    const float* Wgat = (const float*)d_in[7];  // (4,64,64)
    const float* agat = (const float*)d_in[8];  // (4,128)

    float* vout = (float*)d_out;            // v_tilde: 1024*64
    float* pout = (float*)d_out + Nn * 64;  // p: 1024*1024*4

    float* ws = (float*)d_ws;
    float* Cws  = ws + 0;       // 4*32
    float* MVP  = ws + 128;     // 32*4
    float* Aws  = ws + 256;     // 1024*32
    float* Bws  = ws + 33024;   // 1024*32
    float* f1   = ws + 65792;   // 4*1024
    float* f2   = ws + 69888;   // 4*1024
    float* rmax = ws + 73984;   // 4*1024
    float* rinv = ws + 78080;   // 4*1024
    float* hall = ws + 82176;   // 4*1024*64
    unsigned char* maskb = (unsigned char*)(ws + 344320);  // 1024*1024 bytes

    scgrn_k1_small<<<1, 256, 0, stream>>>(Wq, mk, mv, Wp_w, Cws, MVP);
    scgrn_k2_node<<<Nn, 256, 0, stream>>>(nf, Wq, mk, Wgat, agat, Aws, Bws, hall, f1, f2);
    scgrn_k3_pairs<<<dim3(4, Nn), 256, 0, stream>>>(R, Aws, Bws, Cws, MVP, Wp_b, pout, maskb);
    scgrn_k4_rowred<<<Nn, 256, 0, stream>>>(maskb, f1, f2, rmax, rinv);
    scgrn_k5_vgemm<<<256, 32, 0, stream>>>(pout, hall, f1, f2, rmax, rinv, vout);
}